// AttentionLayer_37976100831584
// MI455X (gfx1250) — compile-verified
//
#include <hip/hip_runtime.h>
#include <hip/hip_bf16.h>

#define BB 32
#define TT 2048
#define DD 1024

typedef __attribute__((ext_vector_type(2))) float v2f;
typedef __attribute__((ext_vector_type(8))) float v8f;

// ---------------------------------------------------------------------------
// Init: zero per-batch softmax sums and the output accumulator (both are
// accumulated with global_atomic_add_f32).
// ---------------------------------------------------------------------------
__global__ __launch_bounds__(256) void att_init_kernel(float* __restrict__ sums,
                                                       float* __restrict__ out) {
    int i = blockIdx.x * 256 + threadIdx.x;
    if (i < BB) sums[i] = 0.0f;
    if (i < BB * DD) out[i] = 0.0f;
}

// ---------------------------------------------------------------------------
// Pass 1: scores. Each wave computes 16 t-scores via WMMA f32 16x16x4,
// reducing over D=1024 in K=4 chunks.
//
//   A (16x4, broadcast): A[m,k] = w[k0+k] for all m
//       lane L<16  -> {w[k0],   w[k0+1]}   (float2)
//       lane L>=16 -> {w[k0+2], w[k0+3]}   (float2)
//   B (4x16, streamed):  B[k,n] = x[t0+n, k0+k]
//       lane L<16  -> {x[t0+L, k0],   x[t0+L, k0+1]}   (float2)
//       lane L>=16 -> {x[t0+L', k0+2], x[t0+L', k0+3]} (float2)
//
// C[m,n] = score[t0+n] in every row m -> lane n owns score t0+(n&15), so the
// tanh/exp epilogue runs 16-wide, the e-store is coalesced, and the softmax
// denominator is a shfl_xor reduction + one atomicAdd per wave.
// ---------------------------------------------------------------------------
__global__ __launch_bounds__(256) void att_score_kernel(const float* __restrict__ x,
                                                        const float* __restrict__ w,
                                                        const float* __restrict__ bias,
                                                        float* __restrict__ e,
                                                        float* __restrict__ sums) {
    const int lane  = threadIdx.x & 31;
    const int wave  = threadIdx.x >> 5;
    const int b     = blockIdx.x >> 4;          // 16 blocks per batch (T/128)
    const int tblk  = blockIdx.x & 15;
    const int t0    = tblk * 128 + wave * 16;   // this wave's 16 t-rows
    const int laneM = lane & 15;
    const int kofs  = (lane >> 4) << 1;         // 0 for lanes 0-15, 2 for 16-31

    const float* xrow = x + ((size_t)b * TT + (size_t)(t0 + laneM)) * DD + kofs;
    const float* wp   = w + kofs;

    v8f acc = {};
    #pragma unroll 4
    for (int k = 0; k < DD; k += 4) {
        v2f av = *(const v2f*)(wp + k);         // A: broadcast weight chunk
        v2f bv = *(const v2f*)(xrow + k);       // B: 16 input rows
        acc = __builtin_amdgcn_wmma_f32_16x16x4_f32(
            /*neg_a=*/false, av, /*neg_b=*/false, bv,
            /*c_mod=*/(short)0, acc, /*reuse_a=*/false, /*reuse_b=*/false);
    }

    // Every lane: score for t0 + (lane&15) (lanes 16-31 duplicate 0-15).
    const int t = t0 + laneM;
    const float s  = acc[0] + bias[t];
    // overflow-safe tanh: 1 - 2*rcp(exp(2s)+1)  -> exact +/-1 at extremes
    const float th = 1.0f - 2.0f * __builtin_amdgcn_rcpf(__expf(2.0f * s) + 1.0f);
    const float ev = __expf(th);                // th in [-1,1]: no max needed

    if (lane < 16) e[(size_t)b * TT + t] = ev;  // coalesced store

    // Reduce ev over the 16-lane group (halves hold duplicates, so xor masks
    // 1,2,4,8 give the full 16-term sum in every lane).
    float psum = ev;
    psum += __shfl_xor(psum, 1, 32);
    psum += __shfl_xor(psum, 2, 32);
    psum += __shfl_xor(psum, 4, 32);
    psum += __shfl_xor(psum, 8, 32);
    if (lane == 0) atomicAdd(&sums[b], psum);
}

// ---------------------------------------------------------------------------
// Pass 2: weighted-sum pooling. Each wave owns 16 d-columns of one batch and
// reduces a T/4 slice over t in K=4 chunks via WMMA f32 16x16x4.
//
//   A (16x4, broadcast): A[m,k] = e[t+k] for all m        (float2 load)
//   B (4x16, streamed):  B[k,n] = x[t+k, d0+n]
//       lane L<16  -> {x[t+0, d0+L],  x[t+1, d0+L]}
//       lane L>=16 -> {x[t+2, d0+L'], x[t+3, d0+L']}
//
// C[m,n] = partial out[d0+n] in every row -> lane n holds out[d0+n]; one
// coalesced 16-lane atomicAdd merges the 4 T-slices.
// ---------------------------------------------------------------------------
__global__ __launch_bounds__(256) void att_pool_kernel(const float* __restrict__ x,
                                                       const float* __restrict__ e,
                                                       const float* __restrict__ sums,
                                                       float* __restrict__ out) {
    const int lane   = threadIdx.x & 31;
    const int wave   = threadIdx.x >> 5;
    const int bid    = blockIdx.x;
    const int tslice = bid & 3;                 // 4 T-slices
    const int dtile  = (bid >> 2) & 7;          // 8 d-tiles of 128
    const int b      = bid >> 5;
    const int d0     = dtile * 128 + wave * 16; // this wave's 16 d-columns
    const int laneM  = lane & 15;
    const int tofs   = (lane >> 4) << 1;        // 0 or 2

    const float* xbase = x + (size_t)b * TT * DD + (size_t)d0 + laneM;
    const float* ebase = e + (size_t)b * TT + tofs;

    const int tstart = tslice * (TT / 4);
    v8f acc = {};
    #pragma unroll 4
    for (int t = tstart; t < tstart + TT / 4; t += 4) {
        const int tt = t + tofs;
        v2f av = *(const v2f*)(ebase + t);      // A: broadcast softmax weights
        v2f bv;                                 // B: 16 d-columns, 2 t-rows
        bv.x = xbase[(size_t)tt * DD];
        bv.y = xbase[(size_t)(tt + 1) * DD];
        acc = __builtin_amdgcn_wmma_f32_16x16x4_f32(
            false, av, false, bv, (short)0, acc, false, false);
    }

    if (lane < 16) {
        const float inv = __builtin_amdgcn_rcpf(sums[b]);
        atomicAdd(&out[(size_t)b * DD + d0 + lane], acc[0] * inv);
    }
}

// ---------------------------------------------------------------------------
// Launch
// ---------------------------------------------------------------------------
extern "C" void kernel_launch(void* const* d_in, const int* in_sizes, int n_in,
                              void* d_out, int out_size, void* d_ws, size_t ws_size,
                              hipStream_t stream) {
    const float* x    = (const float*)d_in[0];   // [B,T,D] fp32
    const float* w    = (const float*)d_in[1];   // [D,1]   fp32
    const float* bias = (const float*)d_in[2];   // [T,1]   fp32
    float* out = (float*)d_out;                  // [B,D]   fp32

    float* e    = (float*)d_ws;                  // [B,T] softmax numerators
    float* sums = e + (size_t)BB * TT;           // [B]   softmax denominators

    att_init_kernel<<<(BB * DD + 255) / 256, 256, 0, stream>>>(sums, out);
    att_score_kernel<<<BB * (TT / 128), 256, 0, stream>>>(x, w, bias, e, sums);
    att_pool_kernel<<<BB * 8 * 4, 256, 0, stream>>>(x, e, sums, out);
}